// GCN2_35716948034101
// MI455X (gfx1250) — compile-verified
//
#include <hip/hip_runtime.h>
#include <math.h>

typedef __attribute__((ext_vector_type(2))) float v2f;
typedef __attribute__((ext_vector_type(8))) float v8f;

#define NC 32
#define ALPHA 0.1f

// ---------------- degree / norm ----------------
__global__ void gcn2_deg_init(float* deg, int n) {
    int i = blockIdx.x * blockDim.x + threadIdx.x;
    if (i < n) deg[i] = 1.0f;  // self-loop contribution
}

__global__ void gcn2_deg_count(const int* __restrict__ dst, float* deg, int e) {
    int i = blockIdx.x * blockDim.x + threadIdx.x;
    if (i < e) atomicAdd(&deg[dst[i]], 1.0f);
}

__global__ void gcn2_deg_inv(float* deg, int n) {
    int i = blockIdx.x * blockDim.x + threadIdx.x;
    if (i < n) {
        float d = deg[i];
        deg[i] = (d > 0.0f) ? (1.0f / sqrtf(d)) : 0.0f;
    }
}

// ---------------- stem: h = x @ sw^T + sb ; h0 = h ----------------
__global__ void gcn2_stem(const float* __restrict__ x, const float* __restrict__ sw,
                          const float* __restrict__ sb, float* __restrict__ h,
                          float* __restrict__ h0, int n) {
    int t = blockIdx.x * blockDim.x + threadIdx.x;
    int i = t >> 5;
    int c = t & 31;
    if (i < n) {
        float v = x[i] * sw[c] + sb[c];
        h[i * NC + c] = v;
        h0[i * NC + c] = v;
    }
}

// ---------------- agg init: self-loop term (also zero-fills) ----------------
__global__ void gcn2_init_agg(const float* __restrict__ dinv, const float* __restrict__ h,
                              float* __restrict__ agg, int n) {
    int t = blockIdx.x * blockDim.x + threadIdx.x;
    int i = t >> 5;
    int c = t & 31;
    if (i < n) {
        float dv = dinv[i];
        agg[i * NC + c] = dv * dv * h[i * NC + c];
    }
}

// ---------------- sparse scatter: agg[dst] += norm * h[src] ----------------
// one wave per edge, lane == channel -> 128B coalesced gather + coalesced f32 atomics
__global__ void gcn2_scatter(const int* __restrict__ src, const int* __restrict__ dst,
                             const float* __restrict__ dinv, const float* __restrict__ h,
                             float* __restrict__ agg, int e) {
    int t = blockIdx.x * blockDim.x + threadIdx.x;
    int edge = t >> 5;
    int c = t & 31;
    if (edge < e) {
        int s = src[edge];
        int d = dst[edge];
        float nv = dinv[s] * dinv[d];
        atomicAdd(&agg[d * NC + c], nv * h[s * NC + c]);
    }
}

// ---------------- combine + fp32 WMMA: h = (1-b)*s + b*(s @ w) ----------------
// s = (1-alpha)*agg + alpha*h0, staged in LDS.
// One wave computes a 16x32 output slab via V_WMMA_F32_16X16X4_F32 (8 K-steps x 2 N-tiles).
__global__ void gcn2_combine_wmma(const float* __restrict__ agg, const float* __restrict__ h0,
                                  const float* __restrict__ w, float* __restrict__ h,
                                  float beta, int n) {
    __shared__ float s_tile[128 * NC];  // 16 KB: 128 rows per 256-thread block

    int lane = threadIdx.x & 31;
    int wave = threadIdx.x >> 5;      // 0..7
    int row0 = blockIdx.x * 128;

    // stage s into LDS (clamped rows keep EXEC uniform for WMMA)
    for (int i = threadIdx.x; i < 128 * NC; i += 256) {
        int r = i >> 5;
        int c = i & 31;
        int gr = row0 + r;
        int grc = (gr < n) ? gr : (n - 1);
        s_tile[i] = (1.0f - ALPHA) * agg[grc * NC + c] + ALPHA * h0[grc * NC + c];
    }
    __syncthreads();

    int m0 = wave * 16;               // 16-row tile within LDS slab
    int half = lane >> 4;             // 0: K pair {k,k+1}; 1: K pair {k+2,k+3}
    int l16 = lane & 15;

    v8f acc0 = {};                    // output cols 0..15
    v8f acc1 = {};                    // output cols 16..31

    for (int kk = 0; kk < NC; kk += 4) {
        int k0 = kk + half * 2;
        // A fragment: 16x4 f32 (2 VGPRs): lanes 0-15 hold K={kk,kk+1}, 16-31 hold K={kk+2,kk+3}
        v2f a;
        a.x = s_tile[(m0 + l16) * NC + k0];
        a.y = s_tile[(m0 + l16) * NC + k0 + 1];
        // B fragments: 4x16 f32, row-per-halfwave mirroring A's K split
        v2f b0, b1;
        b0.x = w[k0 * NC + l16];
        b0.y = w[(k0 + 1) * NC + l16];
        b1.x = w[k0 * NC + 16 + l16];
        b1.y = w[(k0 + 1) * NC + 16 + l16];
        acc0 = __builtin_amdgcn_wmma_f32_16x16x4_f32(false, a, false, b0, (short)0, acc0,
                                                     false, false);
        acc1 = __builtin_amdgcn_wmma_f32_16x16x4_f32(false, a, false, b1, (short)0, acc1,
                                                     false, false);
    }

    // C/D layout: VGPR r -> row m0+r (lanes 0-15) / m0+r+8 (lanes 16-31), lane%16 = col
    float omb = 1.0f - beta;
    for (int r = 0; r < 8; ++r) {
        int mr = m0 + r + half * 8;
        int gr = row0 + mr;
        if (gr < n) {
            float s0 = s_tile[mr * NC + l16];
            float s1 = s_tile[mr * NC + 16 + l16];
            h[gr * NC + l16]      = omb * s0 + beta * acc0[r];
            h[gr * NC + 16 + l16] = omb * s1 + beta * acc1[r];
        }
    }
}

// ---------------- head: out = h @ lin_w^T + lin_b ----------------
__global__ void gcn2_head(const float* __restrict__ h, const float* __restrict__ lw,
                          const float* __restrict__ lb, float* __restrict__ out, int n) {
    int i = blockIdx.x * blockDim.x + threadIdx.x;
    if (i < n) {
        float acc = lb[0];
#pragma unroll
        for (int c = 0; c < NC; ++c) acc += h[i * NC + c] * lw[c];
        out[i] = acc;
    }
}

extern "C" void kernel_launch(void* const* d_in, const int* in_sizes, int n_in,
                              void* d_out, int out_size, void* d_ws, size_t ws_size,
                              hipStream_t stream) {
    const float* x   = (const float*)d_in[0];
    const int*   ei  = (const int*)d_in[1];
    const float* sw  = (const float*)d_in[2];
    const float* sb  = (const float*)d_in[3];
    const float* wts = (const float*)d_in[4];
    const float* lw  = (const float*)d_in[5];
    const float* lb  = (const float*)d_in[6];
    float* out = (float*)d_out;

    const int n  = in_sizes[0];            // N nodes (x is [N,1])
    const int e  = in_sizes[1] / 2;        // edges
    const int nb = in_sizes[4] / (NC * NC);
    const int* src = ei;
    const int* dst = ei + e;

    // workspace layout
    float* deg = (float*)d_ws;             // n   (becomes dinv)
    float* h   = deg + n;                  // n*NC
    float* h0  = h + (size_t)n * NC;       // n*NC
    float* agg = h0 + (size_t)n * NC;      // n*NC

    const int B = 256;
    dim3 blk(B);
    int g_n     = (n + B - 1) / B;
    int g_e     = (e + B - 1) / B;
    int g_nc    = ((size_t)n * NC + B - 1) / B;
    int g_ec    = (int)(((size_t)e * NC + B - 1) / B);
    int g_tiles = (n + 127) / 128;

    gcn2_deg_init<<<g_n, blk, 0, stream>>>(deg, n);
    gcn2_deg_count<<<g_e, blk, 0, stream>>>(dst, deg, e);
    gcn2_deg_inv<<<g_n, blk, 0, stream>>>(deg, n);
    gcn2_stem<<<g_nc, blk, 0, stream>>>(x, sw, sb, h, h0, n);

    for (int l = 0; l < nb; ++l) {
        float beta = logf(0.5f / (float)(l + 1) + 1.0f);
        gcn2_init_agg<<<g_nc, blk, 0, stream>>>(deg, h, agg, n);
        gcn2_scatter<<<g_ec, blk, 0, stream>>>(src, dst, deg, h, agg, e);
        gcn2_combine_wmma<<<g_tiles, blk, 0, stream>>>(agg, h0, wts + (size_t)l * NC * NC,
                                                       h, beta, n);
    }

    gcn2_head<<<g_n, blk, 0, stream>>>(h, lw, lb, out, n);
}